// SaliencyGNN_60043642798828
// MI455X (gfx1250) — compile-verified
//
#include <hip/hip_runtime.h>
#include <hip/hip_bf16.h>
#include <stddef.h>

typedef __attribute__((ext_vector_type(2))) float v2f;
typedef __attribute__((ext_vector_type(8))) float v8f;

#define FDIM 64

// ---------------- degree / norm ----------------
__global__ void k_deg_init(float* deg, int n) {
    int i = blockIdx.x * blockDim.x + threadIdx.x;
    if (i < n) deg[i] = 1.0f;  // self loop
}

__global__ void k_deg_count(const long long* __restrict__ dst, float* deg, int e) {
    int i = blockIdx.x * blockDim.x + threadIdx.x;
    if (i < e) atomicAdd(&deg[(int)dst[i]], 1.0f);
}

__global__ void k_rsqrt_inplace(float* deg, int n) {
    int i = blockIdx.x * blockDim.x + threadIdx.x;
    if (i < n) deg[i] = rsqrtf(deg[i]);  // deg >= 1 always
}

// ---------------- GEMM via V_WMMA_F32_16X16X4_F32 ----------------
// One wave computes a 16x16 tile of (X @ W), scales rows by dis, and writes
// both y and agg (agg initialized with the self-loop contribution y).
__global__ __launch_bounds__(256)
void k_gemm_wmma(const float* __restrict__ X, const float* __restrict__ W,
                 const float* __restrict__ dis, float* __restrict__ y,
                 float* __restrict__ agg, int n) {
    const int wave = (blockIdx.x * blockDim.x + threadIdx.x) >> 5;
    const int lane = threadIdx.x & 31;
    const int mt = wave >> 2;        // 4 col-tiles cover 64 outputs
    const int nt = wave & 3;
    const int m0 = mt << 4;
    const int n0 = nt << 4;
    if (m0 >= n) return;             // wave-uniform

    const int r16  = lane & 15;
    const int half = lane >> 4;

    int arow = m0 + r16;
    if (arow >= n) arow = n - 1;     // clamp (n % 16 == 0 here, safety only)
    const float* __restrict__ xr = X + (size_t)arow * FDIM;

    v8f c = {};
#pragma unroll
    for (int k = 0; k < FDIM; k += 4) {
        const int kk = k + (half << 1);
        v2f a, b;
        // A: 16x4 fragment, lane<16 holds K=kk..kk+1, lane>=16 holds K=kk+2..kk+3 pattern via half
        a[0] = xr[kk];
        a[1] = xr[kk + 1];
        // B: 4x16 fragment, same K mapping, col = n0 + r16  (W is [64,64] row-major)
        b[0] = W[(size_t)kk * FDIM + n0 + r16];
        b[1] = W[(size_t)(kk + 1) * FDIM + n0 + r16];
        c = __builtin_amdgcn_wmma_f32_16x16x4_f32(
                /*neg_a=*/false, a, /*neg_b=*/false, b,
                /*c_mod=*/(short)0, c, /*reuse_a=*/false, /*reuse_b=*/false);
    }

#pragma unroll
    for (int v = 0; v < 8; ++v) {
        const int r = m0 + v + (half << 3);   // C/D layout: row = v + 8*(lane>=16)
        if (r < n) {
            const float val = dis[r] * c[v];
            const size_t idx = (size_t)r * FDIM + n0 + r16;
            y[idx]   = val;
            agg[idx] = val;   // self-loop init
        }
    }
}

// ---------------- edge scatter-add (64-wide features) ----------------
// 16 lanes per edge, each moves one float4: one contiguous 256B row per edge.
__global__ __launch_bounds__(256)
void k_scatter64(const long long* __restrict__ src, const long long* __restrict__ dst,
                 const float* __restrict__ y, float* __restrict__ agg, int e) {
    const long long t = (long long)blockIdx.x * blockDim.x + threadIdx.x;
    const int ei = (int)(t >> 4);
    if (ei >= e) return;
    const int q = (int)(t & 15);
    const int s = (int)src[ei];
    const int d = (int)dst[ei];
    const float4 v = *(const float4*)(y + (size_t)s * FDIM + q * 4);
    float* o = agg + (size_t)d * FDIM + q * 4;
    atomicAdd(o + 0, v.x);
    atomicAdd(o + 1, v.y);
    atomicAdd(o + 2, v.z);
    atomicAdd(o + 3, v.w);
}

// ---------------- h = act(dis * agg + b) ----------------
__global__ void k_combine(const float* __restrict__ agg, const float* __restrict__ dis,
                          const float* __restrict__ bias, float* __restrict__ h,
                          int n64, int doRelu) {
    const int i = blockIdx.x * blockDim.x + threadIdx.x;
    if (i >= n64) return;
    const int r = i >> 6;
    const int f = i & 63;
    float v = fmaf(dis[r], agg[i], bias[f]);
    h[i] = doRelu ? fmaxf(v, 0.0f) : v;
}

// ---------------- layer 3: per-node dot with W3 (64x1) ----------------
__global__ void k_gemv_out(const float* __restrict__ h, const float* __restrict__ W3,
                           const float* __restrict__ dis, float* __restrict__ y3,
                           float* __restrict__ agg3, int n) {
    const int i = blockIdx.x * blockDim.x + threadIdx.x;
    if (i >= n) return;
    const float4* hr = (const float4*)(h + (size_t)i * FDIM);
    const float4* w  = (const float4*)W3;
    float acc = 0.0f;
#pragma unroll
    for (int k = 0; k < FDIM / 4; ++k) {
        const float4 a = hr[k];
        const float4 b = w[k];
        acc += a.x * b.x + a.y * b.y + a.z * b.z + a.w * b.w;
    }
    const float v = dis[i] * acc;
    y3[i]   = v;
    agg3[i] = v;  // self loop
}

__global__ void k_scatter1(const long long* __restrict__ src, const long long* __restrict__ dst,
                           const float* __restrict__ y3, float* __restrict__ agg3, int e) {
    const int i = blockIdx.x * blockDim.x + threadIdx.x;
    if (i < e) atomicAdd(&agg3[(int)dst[i]], y3[(int)src[i]]);
}

__global__ void k_final(const float* __restrict__ agg3, const float* __restrict__ dis,
                        const float* __restrict__ b3, float* __restrict__ out, int n) {
    const int i = blockIdx.x * blockDim.x + threadIdx.x;
    if (i < n) out[i] = fmaf(dis[i], agg3[i], b3[0]);
}

extern "C" void kernel_launch(void* const* d_in, const int* in_sizes, int n_in,
                              void* d_out, int out_size, void* d_ws, size_t ws_size,
                              hipStream_t stream) {
    const float*     x  = (const float*)d_in[0];
    const long long* ei = (const long long*)d_in[1];   // edge_index int64 [2, E]
    // d_in[2] = edge_attr (unused by reference)
    const float* W1 = (const float*)d_in[3];
    const float* b1 = (const float*)d_in[4];
    const float* W2 = (const float*)d_in[5];
    const float* b2 = (const float*)d_in[6];
    const float* W3 = (const float*)d_in[7];
    const float* b3 = (const float*)d_in[8];
    float* out = (float*)d_out;

    const int N = in_sizes[0] / FDIM;   // 100000
    const int E = in_sizes[1] / 2;      // 1600000
    const long long* src = ei;
    const long long* dst = ei + E;

    // workspace layout
    char* ws = (char*)d_ws;
    size_t off = 0;
    auto take = [&](size_t bytes) -> void* {
        void* p = ws + off;
        off = (off + bytes + 255) & ~(size_t)255;
        return p;
    };
    float* dis  = (float*)take((size_t)N * 4);            // deg -> dis, in place
    float* bufY = (float*)take((size_t)N * FDIM * 4);     // y
    float* bufA = (float*)take((size_t)N * FDIM * 4);     // agg
    float* bufH = (float*)take((size_t)N * FDIM * 4);     // h
    float* y3   = (float*)take((size_t)N * 4);
    float* agg3 = (float*)take((size_t)N * 4);

    const int T = 256;
    const int gN   = (N + T - 1) / T;
    const int gE   = (E + T - 1) / T;
    const int gN64 = (N * FDIM + T - 1) / T;
    const int gE16 = (int)(((long long)E * 16 + T - 1) / T);
    const int mTiles = (N + 15) / 16;
    const int gGemm  = (mTiles * 4 + 7) / 8;  // 8 waves per block, 1 tile per wave

    // degrees / norm (edge structure is layer-invariant)
    k_deg_init<<<gN, T, 0, stream>>>(dis, N);
    k_deg_count<<<gE, T, 0, stream>>>(dst, dis, E);
    k_rsqrt_inplace<<<gN, T, 0, stream>>>(dis, N);

    // ---- layer 1 ----
    k_gemm_wmma<<<gGemm, T, 0, stream>>>(x, W1, dis, bufY, bufA, N);
    k_scatter64<<<gE16, T, 0, stream>>>(src, dst, bufY, bufA, E);
    k_combine<<<gN64, T, 0, stream>>>(bufA, dis, b1, bufH, N * FDIM, 1);

    // ---- layer 2 ----
    k_gemm_wmma<<<gGemm, T, 0, stream>>>(bufH, W2, dis, bufY, bufA, N);
    k_scatter64<<<gE16, T, 0, stream>>>(src, dst, bufY, bufA, E);
    k_combine<<<gN64, T, 0, stream>>>(bufA, dis, b2, bufH, N * FDIM, 1);

    // ---- layer 3 (out_dim = 1) ----
    k_gemv_out<<<gN, T, 0, stream>>>(bufH, W3, dis, y3, agg3, N);
    k_scatter1<<<gE, T, 0, stream>>>(src, dst, y3, agg3, E);
    k_final<<<gN, T, 0, stream>>>(agg3, dis, b3, out, N);
}